// NCA_23622320128683
// MI455X (gfx1250) — compile-verified
//
#include <hip/hip_runtime.h>
#include <math.h>

// NCA: x <- tanh(x + alpha * relu(circular_conv21(x))), 16 steps, trajectory out.
// alpha = sum_c w3[c]*w2[c]*(w2[c]>0)  (exact collapse of the 1->10->1 ReLU MLP).
// Conv mapped to V_WMMA_F32_16X16X4_F32 via Toeplitz factorization.
// Input tile window staged to LDS via global_load_async_to_lds_b32 (ASYNCcnt).

#define H 512
#define W 512
#define MASKW 511
#define KS 21
#define PADR 10
#define NB 8
#define STEPS 16
#define SLOTS (STEPS + 1)
#define HW (H * W)             // 2^18
#define TILE_N 16
#define MT 20                  // output rows per workgroup
#define RWIN (MT + KS - 1)     // 40-row window
#define XCOLS 40               // staged columns: n(16) + kx(21) + pad taps(3)
#define XSTRIDE 48             // LDS row stride (floats), bank spread
#define NTHREADS 256
#define NWAVES (NTHREADS / 32)

typedef float v2f __attribute__((ext_vector_type(2)));
typedef float v8f __attribute__((ext_vector_type(8)));

__global__ void nca_alpha_kernel(const float* __restrict__ w2,
                                 const float* __restrict__ w3,
                                 float* __restrict__ alpha_out) {
    float a = 0.f;
    for (int c = 0; c < 10; ++c) {
        float s = w2[c];
        if (s > 0.f) a += w3[c] * s;
    }
    alpha_out[0] = a;
}

__global__ void nca_seed_kernel(const float* __restrict__ x,
                                float* __restrict__ traj) {
    int idx = blockIdx.x * blockDim.x + threadIdx.x;
    if (idx < NB * HW) {
        int b = idx >> 18;          // / HW
        int i = idx & (HW - 1);     // % HW
        traj[((size_t)b * SLOTS) * (size_t)HW + i] = x[idx];
    }
}

__launch_bounds__(NTHREADS, 2)
__global__ void nca_step_kernel(float* __restrict__ traj,
                                const float* __restrict__ w1,
                                const float* __restrict__ alpha_p,
                                int step) {
    // G[rw][ky][n]: 40 x 21 x 16 f32 = 53760 B
    __shared__ float Gs[RWIN * KS * TILE_N];
    // Staged input window: 40 rows x 48 f32 = 7680 B  (total 61440 B < 64 KB)
    __shared__ float Xs[RWIN * XSTRIDE];

    const int b  = blockIdx.z;
    const int M0 = blockIdx.y * MT;
    const int C0 = blockIdx.x * TILE_N;

    const float* fin  = traj + ((size_t)b * SLOTS + step) * (size_t)HW;
    float*       fout = traj + ((size_t)b * SLOTS + step + 1) * (size_t)HW;

    const int tid  = threadIdx.x;
    const int lane = tid & 31;          // wave32
    const int wave = tid >> 5;
    const int hi   = (lane >= 16) ? 1 : 0;
    const int n16  = lane & 15;

    // ---- Stage the (RWIN x XCOLS) circularly-wrapped input window into LDS
    // with per-lane async copies (each lane provides wrapped gaddr + LDS offset).
    for (int e = tid; e < RWIN * XCOLS; e += NTHREADS) {
        int rw  = e / XCOLS;
        int j   = e - rw * XCOLS;
        int s   = (M0 - PADR + rw) & MASKW;
        int col = (C0 - PADR + j) & MASKW;
        const float* gp = fin + ((size_t)s << 9) + col;
        unsigned ldsoff = (unsigned)(size_t)&Xs[rw * XSTRIDE + j];
        asm volatile("global_load_async_to_lds_b32 %0, %1, off"
                     :: "v"(ldsoff), "v"(gp) : "memory");
    }
#if __has_builtin(__builtin_amdgcn_s_wait_asynccnt)
    __builtin_amdgcn_s_wait_asynccnt(0);
#else
    asm volatile("s_wait_asynccnt 0" ::: "memory");
#endif
    __syncthreads();

    // ---- Build 12 A-matrices (2 ky-chunks x 6 kx-groups), 16x4 f32 each.
    // ISA 7.12.2 32-bit A 16x4 layout: lanes 0-15: M=lane, {v0=K0, v1=K1};
    // lanes 16-31: M=lane-16, {v0=K2, v1=K3}. Zero-pad ky>=21, kx>=21.
    v2f A[2][6];
#pragma unroll
    for (int c = 0; c < 2; ++c) {
#pragma unroll
        for (int g = 0; g < 6; ++g) {
            int ky = c * 16 + n16;
            int kx = g * 4 + (hi ? 2 : 0);
            float ax = (ky < KS && kx     < KS) ? w1[ky * KS + kx    ] : 0.f;
            float ay = (ky < KS && kx + 1 < KS) ? w1[ky * KS + kx + 1] : 0.f;
            v2f a; a.x = ax; a.y = ay;
            A[c][g] = a;
        }
    }

    // ---- Phase 1: per staged row, 12 chained WMMAs -> G[rw][ky][0..15].
    // B(4x16) Toeplitz slice from LDS: B[k][n] = Xs[rw][n + kx0 + k].
    for (int rw = wave; rw < RWIN; rw += NWAVES) {
        const float* xrow = &Xs[rw * XSTRIDE];
        v8f acc0 = {0.f, 0.f, 0.f, 0.f, 0.f, 0.f, 0.f, 0.f};
        v8f acc1 = {0.f, 0.f, 0.f, 0.f, 0.f, 0.f, 0.f, 0.f};
#pragma unroll
        for (int g = 0; g < 6; ++g) {
            int j = n16 + g * 4 + (hi ? 2 : 0);     // 0..38 < XCOLS
            v2f Bg; Bg.x = xrow[j]; Bg.y = xrow[j + 1];
            acc0 = __builtin_amdgcn_wmma_f32_16x16x4_f32(
                false, A[0][g], false, Bg, (short)0, acc0, false, false);
            acc1 = __builtin_amdgcn_wmma_f32_16x16x4_f32(
                false, A[1][g], false, Bg, (short)0, acc1, false, false);
        }
        // Scatter D -> LDS. D layout: VGPR v, lanes 0-15: M=v; lanes 16-31: M=v+8.
#pragma unroll
        for (int v = 0; v < 8; ++v) {
            int ky = v + (hi ? 8 : 0);                 // chunk 0: ky 0..15
            Gs[(rw * KS + ky) * TILE_N + n16] = acc0[v];
        }
        if (!hi) {
#pragma unroll
            for (int v = 0; v < 5; ++v) {              // chunk 1: ky 16..20 (rest zero)
                Gs[(rw * KS + 16 + v) * TILE_N + n16] = acc1[v];
            }
        }
    }
    __syncthreads();

    // ---- Phase 2: out[m][n] = tanh(x + alpha*relu(sum_ky G[ml+ky][ky][n]))
    const float alpha = alpha_p[0];
    for (int p = tid; p < MT * TILE_N; p += NTHREADS) {
        int ml = p >> 4;
        int nn = p & 15;
        int m  = M0 + ml;
        if (m < H) {
            float acc = 0.f;
#pragma unroll
            for (int ky = 0; ky < KS; ++ky) {
                acc += Gs[((ml + ky) * KS + ky) * TILE_N + nn];
            }
            float xv = Xs[(ml + PADR) * XSTRIDE + nn + PADR];   // center value
            float z  = alpha * fmaxf(acc, 0.f);
            fout[(size_t)m * W + (C0 + nn)] = tanhf(xv + z);
        }
    }
}

extern "C" void kernel_launch(void* const* d_in, const int* in_sizes, int n_in,
                              void* d_out, int out_size, void* d_ws, size_t ws_size,
                              hipStream_t stream) {
    (void)in_sizes; (void)n_in; (void)out_size; (void)ws_size;
    const float* x  = (const float*)d_in[0];
    const float* w1 = (const float*)d_in[1];   // (1,1,21,21) -> 441 floats
    const float* w2 = (const float*)d_in[2];   // 10 floats
    const float* w3 = (const float*)d_in[3];   // 10 floats
    float* traj  = (float*)d_out;              // [8,17,1,512,512]
    float* alpha = (float*)d_ws;

    nca_alpha_kernel<<<1, 1, 0, stream>>>(w2, w3, alpha);
    nca_seed_kernel<<<(NB * HW + 255) / 256, 256, 0, stream>>>(x, traj);

    dim3 grid(W / TILE_N, (H + MT - 1) / MT, NB);   // 32 x 26 x 8
    for (int t = 0; t < STEPS; ++t) {
        nca_step_kernel<<<grid, NTHREADS, 0, stream>>>(traj, w1, alpha, t);
    }
}